// EUNN_66254165508959
// MI455X (gfx1250) — compile-verified
//
#include <hip/hip_runtime.h>

// EUNN: 16 stages of paired complex rotations + roll over hidden=2048, batch=8192.
// Memory-bound (HBM floor ~11.5us on MI455X @ 23.3 TB/s). Strategy: stream rows
// through LDS with CDNA5 async global<->LDS ops, fuse all 16 stages in LDS,
// amortize the per-stage coefficient table across 8 batch rows per block.

#define H    2048   // hidden (complex elements per row)
#define CAP  16     // number of stages
#define NB   8192   // batch
#define TPB  256    // threads per block (8 waves of 32)
#define ROWS 8      // batch rows per block (8 * 16KB = 128KB LDS)
#define PPT  4      // pairs per thread: 1024 pairs / 256 threads

// ---------------------------------------------------------------------------
// Coefficient precompute: ctab[k*1024 + p] = (cos phi, sin phi, cos th, sin th)
// phi = angles[2p, k], th = angles[2p+1, k]; angles is (2048, 16) row-major.
// ---------------------------------------------------------------------------
__global__ __launch_bounds__(TPB)
void eunn_coeff(const float* __restrict__ angles, float4* __restrict__ ctab) {
    int idx = blockIdx.x * TPB + threadIdx.x;       // 0 .. 16*1024-1
    int k = idx >> 10;
    int p = idx & 1023;
    float phi = angles[(2 * p) * CAP + k];
    float th  = angles[(2 * p + 1) * CAP + k];
    float4 c;
    c.x = cosf(phi); c.y = sinf(phi);
    c.z = cosf(th);  c.w = sinf(th);
    ctab[idx] = c;
}

// ---------------------------------------------------------------------------
// Main kernel: one block = ROWS batch rows resident in LDS for all 16 stages.
// ---------------------------------------------------------------------------
template <bool PRE>
__global__ __launch_bounds__(TPB)
void eunn_main(const float* __restrict__ xin, float* __restrict__ xout,
               const float4* __restrict__ ctab, const float* __restrict__ angles) {
    __shared__ alignas(16) float buf[ROWS][H * 2];   // 8 x 16KB = 128KB

    const int t = threadIdx.x;
    const long long row0 = (long long)blockIdx.x * ROWS;

    // ---- Async load: ROWS rows of 16KB each, 4 x b128 per thread per row ----
    for (int r = 0; r < ROWS; ++r) {
        unsigned long long g =
            (unsigned long long)(const void*)(xin + (row0 + r) * (long long)(H * 2));
        unsigned lbase = (unsigned)(unsigned long long)(void*)&buf[r][0];
#pragma unroll
        for (int it = 0; it < 4; ++it) {
            unsigned off   = (unsigned)(it * TPB + t) * 16u;
            unsigned laddr = lbase + off;
            unsigned long long ga = g + off;
            asm volatile("global_load_async_to_lds_b128 %0, %1, off"
                         :: "v"(laddr), "v"(ga) : "memory");
        }
    }
    asm volatile("s_wait_asynccnt 0" ::: "memory");
    __syncthreads();

    // ---- 16 stages, all in LDS ----
    for (int k = 0; k < CAP; ++k) {
        const int s = (k & 1) ? 1 : -1;   // roll amount this stage

        // Per-thread coefficients for its 4 pairs (shared across all 8 rows).
        float cp[PPT], sp[PPT], ct[PPT], st[PPT];
#pragma unroll
        for (int q = 0; q < PPT; ++q) {
            int p = t + q * TPB;
            if (PRE) {
                float4 c = ctab[k * (H / 2) + p];
                cp[q] = c.x; sp[q] = c.y; ct[q] = c.z; st[q] = c.w;
            } else {
                float phi = angles[(2 * p) * CAP + k];
                float th  = angles[(2 * p + 1) * CAP + k];
                __sincosf(phi, &sp[q], &cp[q]);
                __sincosf(th,  &st[q], &ct[q]);
            }
        }

        // Two groups of 4 rows to bound live registers (~64 VGPRs of y-state).
#pragma unroll
        for (int g = 0; g < 2; ++g) {
            float2 y0[4][PPT], y1[4][PPT];
#pragma unroll
            for (int rr = 0; rr < 4; ++rr) {
                const int r = g * 4 + rr;
                const float4* bp = (const float4*)&buf[r][0];
#pragma unroll
                for (int q = 0; q < PPT; ++q) {
                    int p = t + q * TPB;
                    float4 xv = bp[p];  // (x0.re, x0.im, x1.re, x1.im)
                    // u = ct*x0 - st*x1 ; y0 = e^{i phi} * u ; y1 = ct*x1 + st*x0
                    float ur = ct[q] * xv.x - st[q] * xv.z;
                    float ui = ct[q] * xv.y - st[q] * xv.w;
                    y0[rr][q].x = cp[q] * ur - sp[q] * ui;
                    y0[rr][q].y = cp[q] * ui + sp[q] * ur;
                    y1[rr][q].x = ct[q] * xv.z + st[q] * xv.x;
                    y1[rr][q].y = ct[q] * xv.w + st[q] * xv.y;
                }
            }
            __syncthreads();   // all reads of these rows done
#pragma unroll
            for (int rr = 0; rr < 4; ++rr) {
                const int r = g * 4 + rr;
                float2* wp = (float2*)&buf[r][0];
#pragma unroll
                for (int q = 0; q < PPT; ++q) {
                    int p = t + q * TPB;
                    wp[(2 * p + s) & (H - 1)]     = y0[rr][q];  // roll fused
                    wp[(2 * p + 1 + s) & (H - 1)] = y1[rr][q];
                }
            }
            __syncthreads();   // all writes of these rows done
        }
    }

    // ---- Async store: LDS -> global, 4 x b128 per thread per row ----
    for (int r = 0; r < ROWS; ++r) {
        unsigned long long g =
            (unsigned long long)(void*)(xout + (row0 + r) * (long long)(H * 2));
        unsigned lbase = (unsigned)(unsigned long long)(void*)&buf[r][0];
#pragma unroll
        for (int it = 0; it < 4; ++it) {
            unsigned off   = (unsigned)(it * TPB + t) * 16u;
            unsigned laddr = lbase + off;
            unsigned long long ga = g + off;
            asm volatile("global_store_async_from_lds_b128 %0, %1, off"
                         :: "v"(ga), "v"(laddr) : "memory");
        }
    }
    asm volatile("s_wait_asynccnt 0" ::: "memory");
}

extern "C" void kernel_launch(void* const* d_in, const int* in_sizes, int n_in,
                              void* d_out, int out_size, void* d_ws, size_t ws_size,
                              hipStream_t stream) {
    const float* angles = (const float*)d_in[0];   // (2048, 16) f32
    const float* x      = (const float*)d_in[1];   // (8192, 2048, 2) f32
    float*       out    = (float*)d_out;           // (8192, 2048, 2) f32

    const size_t need = (size_t)CAP * (H / 2) * sizeof(float4);  // 256 KB
    if (d_ws != nullptr && ws_size >= need) {
        float4* ctab = (float4*)d_ws;
        eunn_coeff<<<(CAP * (H / 2)) / TPB, TPB, 0, stream>>>(angles, ctab);
        eunn_main<true><<<NB / ROWS, TPB, 0, stream>>>(x, out, ctab, angles);
    } else {
        eunn_main<false><<<NB / ROWS, TPB, 0, stream>>>(x, out, nullptr, angles);
    }
}